// ELBO_26517128086198
// MI455X (gfx1250) — compile-verified
//
#include <hip/hip_runtime.h>
#include <math.h>

// ---------------------------------------------------------------------------
// Problem constants (match reference)
// ---------------------------------------------------------------------------
static constexpr int    NDIM   = 4096;
static constexpr int    KTAB   = 1001;
static constexpr float  MU_LO  = -5.0f;
static constexpr float  MU_HI  =  5.0f;
static constexpr float  LS_LO  = -4.0f;
static constexpr float  LS_HI  =  2.0f;
static constexpr float  INV_DMU = (KTAB - 1) / (MU_HI - MU_LO);   // 100.0
static constexpr float  INV_DLS = (KTAB - 1) / (LS_HI - LS_LO);   // 166.6667
static constexpr float  S2_MIN  = 1e-4f;                          // 10^LS_LO

typedef float v2f __attribute__((ext_vector_type(2)));
typedef float v8f __attribute__((ext_vector_type(8)));

// ---------------------------------------------------------------------------
// tiny helpers
// ---------------------------------------------------------------------------
__global__ void init_accum_kernel(double* accum) {
    accum[0] = 0.0;   // ell + prior row terms
    accum[1] = 0.0;   // sum log diag(L)  (== 0.5 * logdet)
}

// log C(n, w) LUT for n in [0,14], w in [0,7]  -> 120 entries (pad to 128)
__global__ void lut_kernel(float* __restrict__ lutg) {
    int t = threadIdx.x;
    if (t < 120) {
        int n = t >> 3, w = t & 7;
        lutg[t] = lgammaf((float)n + 1.0f) - lgammaf((float)w + 1.0f)
                - lgammaf((float)(n - w) + 1.0f);
    }
}

// one edge-clamped lerp along x for a fixed (iy, fy) column pair
__device__ __forceinline__ float lerp_x(const float* __restrict__ fs,
                                        float tx, int iy, float fy) {
    tx = fminf(fmaxf(tx, 0.0f), (float)(KTAB - 1));
    int ix = (int)floorf(tx);
    ix = min(ix, KTAB - 2);
    float fx = tx - (float)ix;
    const float* r0 = fs + (size_t)ix * KTAB + iy;
    float v00 = r0[0],     v01 = r0[1];
    float v10 = r0[KTAB],  v11 = r0[KTAB + 1];
    float gx = 1.0f - fx, gy = 1.0f - fy;
    return v00 * gx * gy + v10 * fx * gy + v01 * gx * fy + v11 * fx * fy;
}

// ---------------------------------------------------------------------------
// ell + prior pass: one thread per (j,k) element of the N x N grid.
// Bandwidth-bound: streams Sigma / n / w (~200 MB); fs table, diagonal and
// the transpose reads of Sigma all fit in the 192 MB L2.
// ---------------------------------------------------------------------------
__global__ __launch_bounds__(256) void ell_kernel(
    const float* __restrict__ mu, const float* __restrict__ Sigma,
    const float* __restrict__ fs, const int* __restrict__ nmat,
    const int* __restrict__ wmat, const float* __restrict__ lutg,
    double* __restrict__ accum)
{
    __shared__ float lut[128];
    __shared__ float red[256];
    const int tid = threadIdx.x;
    if (tid < 120) lut[tid] = lutg[tid];
    __syncthreads();

    const int j = blockIdx.y;
    const int k = blockIdx.x * 256 + tid;       // N is a multiple of 256
    const size_t jk = (size_t)j * NDIM + k;

    const float muj = mu[j];
    const float muk = mu[k];
    const float dSj = Sigma[(size_t)j * NDIM + j];
    const float dSk = Sigma[(size_t)k * NDIM + k];
    const float sjk = Sigma[jk];
    const float skj = Sigma[(size_t)k * NDIM + j];
    const int   nn  = nmat[jk];
    const int   ww  = wmat[jk];

    const int ni = min(max(nn, 0), 14);
    const int wi = min(max(ww, 0), 7);
    const float logC = lut[ni * 8 + wi];
    const float nf = (float)nn, wf = (float)ww;

    float term;
    if (j == k) {
        // p = q = 0.5 on the diagonal; fold the per-row prior term in here
        term = logC + 0.5f * nf - 0.5f * (muj * muj + dSj);
    } else {
        const float mud = muj - muk;
        const float s2d = fmaxf(dSj + dSk - sjk - skj, S2_MIN);
        float ty = (log10f(s2d) - LS_LO) * INV_DLS;
        ty = fminf(fmaxf(ty, 0.0f), (float)(KTAB - 1));
        int   iy = min((int)floorf(ty), KTAB - 2);
        float fy = ty - (float)iy;
        const float p = lerp_x(fs, ( mud - MU_LO) * INV_DMU, iy, fy);
        const float q = lerp_x(fs, (-mud - MU_LO) * INV_DMU, iy, fy);
        term = logC + wf * p + (nf - wf) * q;
    }

    red[tid] = term;
    __syncthreads();
    for (int s = 128; s > 0; s >>= 1) {
        if (tid < s) red[tid] += red[tid + s];
        __syncthreads();
    }
    if (tid == 0) atomicAdd(accum, (double)red[0]);
}

// ---------------------------------------------------------------------------
// Blocked Cholesky of the Sigma copy in workspace (for slogdet).
// Two-level blocking: 16-wide micro-steps inside a 128-wide outer panel,
// then one large WMMA GEMM trailing update per outer step.
// ---------------------------------------------------------------------------
__global__ __launch_bounds__(32) void chol_diag_kernel(float* __restrict__ W,
                                                       int k,
                                                       double* __restrict__ logacc)
{
    __shared__ float a[16][17];
    const int tid = threadIdx.x;
    for (int e = tid; e < 256; e += 32) {
        int r = e >> 4, c = e & 15;
        a[r][c] = W[(size_t)(k + r) * NDIM + (k + c)];
    }
    __syncthreads();
    for (int j = 0; j < 16; ++j) {
        if (tid == 0) a[j][j] = sqrtf(fmaxf(a[j][j], 1e-30f));
        __syncthreads();
        const float dj = a[j][j];
        if (tid > j && tid < 16) a[tid][j] /= dj;
        __syncthreads();
        if (tid > j && tid < 16) {
            const float lij = a[tid][j];
            for (int c = j + 1; c <= tid; ++c) a[tid][c] -= lij * a[c][j];
        }
        __syncthreads();
    }
    for (int e = tid; e < 256; e += 32) {
        int r = e >> 4, c = e & 15;
        W[(size_t)(k + r) * NDIM + (k + c)] = a[r][c];
    }
    if (tid == 0) {
        float s = 0.0f;
        for (int j = 0; j < 16; ++j) s += logf(a[j][j]);
        *logacc += (double)s;   // single workgroup, stream-ordered: no race
    }
}

// Panel solve: all rows below the diag block, x * L11^T = a_row.
__global__ __launch_bounds__(128) void trsm_kernel(float* __restrict__ W,
                                                   int k, int rows)
{
    __shared__ float L[256];
    const int tid = threadIdx.x;
    for (int e = tid; e < 256; e += 128)
        L[e] = W[(size_t)(k + (e >> 4)) * NDIM + (k + (e & 15))];
    __syncthreads();
    const int i = blockIdx.x * 128 + tid;
    if (i >= rows) return;
    const size_t row = (size_t)(k + 16 + i) * NDIM + k;
    float x[16];
#pragma unroll
    for (int c = 0; c < 16; ++c) {
        float s = W[row + c];
        for (int t = 0; t < c; ++t) s -= x[t] * L[c * 16 + t];
        x[c] = s / L[c * 16 + c];
    }
#pragma unroll
    for (int c = 0; c < 16; ++c) W[row + c] = x[c];
}

// Within-panel rank-16 update: one wave per 16x16 tile (<=7 column tiles),
// D = C - P_I * P_J^T via four chained v_wmma_f32_16x16x4_f32.
__global__ __launch_bounds__(32) void syrk_wmma_kernel(float* __restrict__ W,
                                                       int k)
{
    const int I = blockIdx.y, J = blockIdx.x;
    if (J > I) return;                     // lower-triangular tiles only
    const int base = k + 16;
    const int lane = threadIdx.x;
    const int half = lane >> 4;
    const int l15  = lane & 15;

    const size_t rowI = (size_t)(base + I * 16);
    const size_t rowJ = (size_t)(base + J * 16);
    const size_t pI = (rowI + l15) * NDIM + k;
    const size_t pJ = (rowJ + l15) * NDIM + k;

    v8f c;
    const int m0 = half * 8;
#pragma unroll
    for (int r = 0; r < 8; ++r)
        c[r] = W[(rowI + m0 + r) * NDIM + rowJ + l15];

#pragma unroll
    for (int kc = 0; kc < 4; ++kc) {
        const int off = kc * 4 + half * 2;
        v2f a, b;
        a[0] = -W[pI + off];     a[1] = -W[pI + off + 1];
        b[0] =  W[pJ + off];     b[1] =  W[pJ + off + 1];
        c = __builtin_amdgcn_wmma_f32_16x16x4_f32(
                false, a, false, b, (short)0, c, false, false);
    }

#pragma unroll
    for (int r = 0; r < 8; ++r)
        W[(rowI + m0 + r) * NDIM + rowJ + l15] = c[r];
}

// Big trailing update, K = 128: C[128x128 macro-tile] -= P_I * P_J^T.
// 256 threads = 8 waves; each wave owns a 32x64 C block = 8 WMMA tiles.
// Panel blocks staged through LDS (K-blocked x32), A pre-negated in LDS.
__global__ __launch_bounds__(256) void gemm_syrk128_kernel(float* __restrict__ W,
                                                           int k0)
{
    const int bI = blockIdx.y, bJ = blockIdx.x;
    if (bJ > bI) return;                   // lower-triangular macro-tiles
    const int base = k0 + 128;

    __shared__ float As[128][36];          // -P rows of I block (36: pad vs banks)
    __shared__ float Bs[128][36];          //  P rows of J block

    const int tid  = threadIdx.x;
    const int lane = tid & 31;
    const int wv   = tid >> 5;
    const int half = lane >> 4;
    const int l15  = lane & 15;
    const int mwb  = (wv & 3) * 32;        // wave row base within macro-tile
    const int nwb  = (wv >> 2) * 64;       // wave col base within macro-tile

    const size_t RI = (size_t)(base + bI * 128);
    const size_t RJ = (size_t)(base + bJ * 128);

    // load C sub-tiles (WMMA f32 C layout: element (M = half*8 + r, N = l15))
    v8f c[2][4];
#pragma unroll
    for (int mt = 0; mt < 2; ++mt) {
        const size_t gr = RI + mwb + mt * 16 + half * 8;
#pragma unroll
        for (int nt = 0; nt < 4; ++nt) {
            const size_t gc = RJ + nwb + nt * 16 + l15;
#pragma unroll
            for (int r = 0; r < 8; ++r)
                c[mt][nt][r] = W[(gr + r) * NDIM + gc];
        }
    }

    const int ldr = tid >> 3;              // 0..31
    const int ldq = (tid & 7) * 4;         // 0,4,...,28
    for (int kb = 0; kb < 4; ++kb) {
        __syncthreads();
        const int kc0 = k0 + kb * 32;
#pragma unroll
        for (int rr = 0; rr < 4; ++rr) {
            const int row = ldr + rr * 32;
            const float4 va = *(const float4*)&W[(RI + row) * NDIM + kc0 + ldq];
            const float4 vb = *(const float4*)&W[(RJ + row) * NDIM + kc0 + ldq];
            As[row][ldq + 0] = -va.x; As[row][ldq + 1] = -va.y;
            As[row][ldq + 2] = -va.z; As[row][ldq + 3] = -va.w;
            Bs[row][ldq + 0] =  vb.x; Bs[row][ldq + 1] =  vb.y;
            Bs[row][ldq + 2] =  vb.z; Bs[row][ldq + 3] =  vb.w;
        }
        __syncthreads();

#pragma unroll
        for (int kk = 0; kk < 32; kk += 4) {
            const int col = kk + half * 2;
            v2f a[2], b[4];
#pragma unroll
            for (int mt = 0; mt < 2; ++mt)
                a[mt] = *(const v2f*)&As[mwb + mt * 16 + l15][col];
#pragma unroll
            for (int nt = 0; nt < 4; ++nt)
                b[nt] = *(const v2f*)&Bs[nwb + nt * 16 + l15][col];
#pragma unroll
            for (int mt = 0; mt < 2; ++mt)
#pragma unroll
                for (int nt = 0; nt < 4; ++nt)
                    c[mt][nt] = __builtin_amdgcn_wmma_f32_16x16x4_f32(
                        false, a[mt], false, b[nt], (short)0, c[mt][nt],
                        false, false);
        }
    }

    // store C back
#pragma unroll
    for (int mt = 0; mt < 2; ++mt) {
        const size_t gr = RI + mwb + mt * 16 + half * 8;
#pragma unroll
        for (int nt = 0; nt < 4; ++nt) {
            const size_t gc = RJ + nwb + nt * 16 + l15;
#pragma unroll
            for (int r = 0; r < 8; ++r)
                W[(gr + r) * NDIM + gc] = c[mt][nt][r];
        }
    }
}

// ---------------------------------------------------------------------------
// finalize: out = ell + prior + entropy
// ---------------------------------------------------------------------------
__global__ void finalize_kernel(const double* __restrict__ accum,
                                double consts, float* __restrict__ out)
{
    // accum[1] = sum log diag(L) = 0.5 * logdet
    out[0] = (float)(accum[0] + consts + accum[1]);
}

// ---------------------------------------------------------------------------
extern "C" void kernel_launch(void* const* d_in, const int* in_sizes, int n_in,
                              void* d_out, int out_size, void* d_ws, size_t ws_size,
                              hipStream_t stream)
{
    (void)in_sizes; (void)n_in; (void)out_size; (void)ws_size;
    const float* mu    = (const float*)d_in[0];
    const float* Sigma = (const float*)d_in[1];
    const float* fs    = (const float*)d_in[2];
    // d_in[3] = dmus, d_in[4] = ds2s : only used by the reference backward
    const int*   nmat  = (const int*)d_in[5];
    const int*   wmat  = (const int*)d_in[6];
    float* out = (float*)d_out;

    char* ws = (char*)d_ws;
    const size_t matBytes = (size_t)NDIM * NDIM * sizeof(float);
    float*  W     = (float*)ws;                         // Sigma copy, factored in place
    float*  lutg  = (float*)(ws + matBytes);            // 128 floats
    double* accum = (double*)(ws + matBytes + 512);     // [ell, sum log diag]

    hipMemcpyAsync(W, Sigma, matBytes, hipMemcpyDeviceToDevice, stream);
    init_accum_kernel<<<1, 1, 0, stream>>>(accum);
    lut_kernel<<<1, 128, 0, stream>>>(lutg);

    dim3 gEll(NDIM / 256, NDIM);
    ell_kernel<<<gEll, 256, 0, stream>>>(mu, Sigma, fs, nmat, wmat, lutg, accum);

    for (int k0 = 0; k0 < NDIM; k0 += 128) {
        // --- factor the 128-wide panel with 16-wide micro-steps ---
        for (int kb2 = 0; kb2 < 8; ++kb2) {
            const int k = k0 + kb2 * 16;
            chol_diag_kernel<<<1, 32, 0, stream>>>(W, k, accum + 1);
            const int rows = NDIM - k - 16;
            if (rows > 0) {
                trsm_kernel<<<(rows + 127) / 128, 128, 0, stream>>>(W, k, rows);
                const int colTiles = 7 - kb2;          // remaining panel columns
                const int rowTiles = rows / 16;
                if (colTiles > 0)
                    syrk_wmma_kernel<<<dim3(colTiles, rowTiles), 32, 0, stream>>>(W, k);
            }
        }
        // --- one large K=128 WMMA GEMM trailing update ---
        const int M = NDIM - k0 - 128;
        if (M > 0) {
            const int TM = M / 128;
            gemm_syrk128_kernel<<<dim3(TM, TM), 256, 0, stream>>>(W, k0);
        }
    }

    const double TWO_PI = 6.283185307179586;
    const double consts =
        -(double)NDIM * (double)NDIM * 0.5 * log(TWO_PI)          // prior const
        + 0.5 * (double)NDIM * (log(TWO_PI) + 1.0);               // entropy const
    finalize_kernel<<<1, 1, 0, stream>>>(accum, consts, out);
}